// QuantizedGroupEmbedding_55585466745294
// MI455X (gfx1250) — compile-verified
//
#include <hip/hip_runtime.h>

// ---------------------------------------------------------------------------
// QuantizedGroupEmbedding for MI455X (gfx1250)
//   out[b,s,:] = (unpack_int4(weight[idx]) - 8) * scales[idx, group]   (fp16)
// Roofline: ~34 MB gathered reads + 128 MiB streamed fp16 writes => ~7 us
// floor at 23.3 TB/s; purely store-bandwidth bound, no matmul (WMMA N/A).
// CDNA5 paths: GLOBAL_LOAD_ASYNC_TO_LDS_B64 double buffering + s_wait_asynccnt,
// packed v_pk_* fp16 dequant, 128-bit non-temporal stores.
// ---------------------------------------------------------------------------

#define EMBED_DIM        4096
#define ROW_PACKED_BYTES 2048   // EMBED_DIM/2 int4 codes
#define GROUPS           16
#define THREADS          256
#define ROWS_PER_BLOCK   4

typedef __attribute__((ext_vector_type(2))) _Float16     h2_t;
typedef __attribute__((ext_vector_type(4))) unsigned int u32x4_t;
typedef __attribute__((ext_vector_type(2))) int          v2i_t;

// Address-space-qualified element types for the async-copy builtin:
// __builtin_amdgcn_global_load_async_to_lds_b64(AS1 v2i*, AS3 v2i*, Ii, Ii)
typedef v2i_t __attribute__((address_space(1))) v2i_g;
typedef v2i_t __attribute__((address_space(3))) v2i_l;

#if __has_builtin(__builtin_amdgcn_global_load_async_to_lds_b64)
#define CDNA5_ASYNC 1
#else
#define CDNA5_ASYNC 0
#endif

static __device__ __forceinline__ h2_t as_h2(unsigned int x) {
  h2_t r; __builtin_memcpy(&r, &x, 4); return r;
}
static __device__ __forceinline__ unsigned int as_u32(h2_t x) {
  unsigned int r; __builtin_memcpy(&r, &x, 4); return r;
}

// Copy 8 bytes global -> LDS for this lane (async on CDNA5).
static __device__ __forceinline__ void async_cp8(const unsigned char* g,
                                                 unsigned char* l) {
#if CDNA5_ASYNC
  __builtin_amdgcn_global_load_async_to_lds_b64(
      (v2i_g*)(void*)g, (v2i_l*)(void*)l, /*offset=*/0, /*cpol=*/0);
#else
  *(unsigned long long*)l = *(const unsigned long long*)g;
#endif
}

template <int N>
static __device__ __forceinline__ void wait_async() {
#if CDNA5_ASYNC
#if __has_builtin(__builtin_amdgcn_s_wait_asynccnt)
  __builtin_amdgcn_s_wait_asynccnt(N);
#else
  asm volatile("s_wait_asynccnt %0" ::"i"(N));
#endif
#endif
}

// Dequant 4 packed bytes -> 4 half2 words (8 fp16 outputs).
// byte b: element 2k = (b>>4)-8, element 2k+1 = (b&0xF)-8, times scale.
// fp16 bit trick: 0x6400|n == (half)(1024+n) exactly; (1024+n)-1032 == n-8
// exactly; then one packed multiply by the group scale (matches reference
// rounding: single fp16 multiply of an exact small-int fp16 by the scale).
static __device__ __forceinline__ void dq_word(unsigned int w, h2_t s2,
                                               unsigned int* outbits) {
  const h2_t bias = as_h2(0x64086408u);  // (1032.0h, 1032.0h)
#pragma unroll
  for (int k = 0; k < 4; ++k) {
    unsigned int b    = (w >> (8 * k)) & 0xFFu;
    unsigned int bits = 0x64006400u | (b >> 4) | ((b & 0xFu) << 16);
    h2_t q      = as_h2(bits) - bias;  // exact (n-8) in both halves
    outbits[k]  = as_u32(q * s2);      // v_pk_mul_f16
  }
}

// Dequant + store one gathered row from an LDS staging buffer.
static __device__ __forceinline__ void process_row(
    int row, long long idx, const unsigned char* lbufrow,
    const _Float16* __restrict__ scales, _Float16* __restrict__ out, int tid) {
  // This thread's 16 output elements [tid*16, tid*16+16) all belong to
  // group tid/16 (groupsize 256 = 16 threads * 16 elements).
  const _Float16 s  = scales[(size_t)idx * GROUPS + (tid >> 4)];
  const h2_t     s2 = {s, s};

  const unsigned int* lp = (const unsigned int*)(lbufrow + tid * 8);  // ds 8B
  const unsigned int w0 = lp[0];
  const unsigned int w1 = lp[1];

  unsigned int ob[8];
  dq_word(w0, s2, ob);
  dq_word(w1, s2, ob + 4);

  // 32 B of fp16 output per thread, contiguous across the block:
  // two 128-bit non-temporal stores (write-once 128 MiB stream).
  _Float16* op = out + (size_t)row * EMBED_DIM + (size_t)tid * 16;
  u32x4_t v0 = {ob[0], ob[1], ob[2], ob[3]};
  u32x4_t v1 = {ob[4], ob[5], ob[6], ob[7]};
  __builtin_nontemporal_store(v0, (u32x4_t*)op);
  __builtin_nontemporal_store(v1, (u32x4_t*)(op + 8));
}

__global__ __launch_bounds__(THREADS) void qembed_dequant_kernel(
    const long long* __restrict__ indices,      // [nrows] int64
    const unsigned char* __restrict__ weight,   // [VOCAB, 2048] packed int4
    const _Float16* __restrict__ scales,        // [VOCAB, 16] fp16
    _Float16* __restrict__ out,                 // [nrows, 4096] fp16
    int nrows) {
  __shared__ unsigned char lbuf[2][ROW_PACKED_BYTES];

  const int tid  = threadIdx.x;
  const int row0 = blockIdx.x * ROWS_PER_BLOCK;
  if (row0 >= nrows) return;

  if (row0 + ROWS_PER_BLOCK <= nrows) {
    // ---- Hot path: full block, branch-free, fully unrolled. ----
    {
      const long long idx0 = indices[row0];
      async_cp8(weight + (size_t)idx0 * ROW_PACKED_BYTES + (size_t)tid * 8,
                &lbuf[0][tid * 8]);
    }
#pragma unroll
    for (int r = 0; r < ROWS_PER_BLOCK; ++r) {
      const int buf = r & 1;
      // Prefetch next row into the other buffer, then wait for the current
      // one. Async loads complete in order per wave => asynccnt<=1 means the
      // current row's copy is done while the next stays in flight.
      if (r + 1 < ROWS_PER_BLOCK) {
        const long long idxn = indices[row0 + r + 1];
        async_cp8(weight + (size_t)idxn * ROW_PACKED_BYTES + (size_t)tid * 8,
                  &lbuf[buf ^ 1][tid * 8]);
        wait_async<1>();
      } else {
        wait_async<0>();
      }
      __syncthreads();  // all 8 waves' LDS slices of this row are complete
      process_row(row0 + r, indices[row0 + r], &lbuf[buf][0], scales, out, tid);
      __syncthreads();  // buffer may be overwritten by next iter's prefetch
    }
  } else {
    // ---- Tail path (only the final partial block, if any). ----
    for (int r = 0; r < ROWS_PER_BLOCK; ++r) {
      const int row = row0 + r;
      if (row >= nrows) break;
      const long long idx = indices[row];
      async_cp8(weight + (size_t)idx * ROW_PACKED_BYTES + (size_t)tid * 8,
                &lbuf[0][tid * 8]);
      wait_async<0>();
      __syncthreads();
      process_row(row, idx, &lbuf[0][0], scales, out, tid);
      __syncthreads();
    }
  }
}

extern "C" void kernel_launch(void* const* d_in, const int* in_sizes, int n_in,
                              void* d_out, int out_size, void* d_ws,
                              size_t ws_size, hipStream_t stream) {
  const long long*     indices = (const long long*)d_in[0];    // int64 [4,4096]
  const unsigned char* weight  = (const unsigned char*)d_in[1];// uint8 [VOCAB,2048]
  const _Float16*      scales  = (const _Float16*)d_in[2];     // fp16  [VOCAB,16]
  _Float16*            out     = (_Float16*)d_out;             // fp16  [4,4096,4096]

  const int nrows  = in_sizes[0];  // 16384 lookups
  const int blocks = (nrows + ROWS_PER_BLOCK - 1) / ROWS_PER_BLOCK;
  qembed_dequant_kernel<<<blocks, THREADS, 0, stream>>>(indices, weight, scales,
                                                        out, nrows);
}